// WassersteinMetric_14998025798407
// MI455X (gfx1250) — compile-verified
//
#include <hip/hip_runtime.h>
#include <cstdint>

#define ROW_N    8192
#define NTHREADS 1024

// One workgroup per row. Stages u-row into the first N floats of LDS and v-row into the next N floats
// via gfx1250 async global->LDS b128 DMA, bitonic-sorts both concurrently,
// then computes (1/N) * sum |u_sorted[i] - v_sorted[i]|.
__global__ __launch_bounds__(NTHREADS)
void WassersteinMetric_14998025798407_kernel(const float* __restrict__ u,
                                             const float* __restrict__ v,
                                             float* __restrict__ out)
{
    __shared__ __align__(16) float smem[2 * ROW_N];   // 64 KB

    const int tid = threadIdx.x;
    const int row = blockIdx.x;

    const float* __restrict__ urow = u + (size_t)row * ROW_N;
    const float* __restrict__ vrow = v + (size_t)row * ROW_N;

    // LDS byte address of smem base (low 32 bits of the flat shared pointer).
    const uint32_t lds_base = (uint32_t)(uintptr_t)(&smem[0]);

    // Async copy: each array is 32 KB = 2048 x b128; 1024 lanes -> 2 passes/array.
    // GVS mode: source is the 64-bit saddr plus the 32-bit offset in vaddr; vdst is the LDS byte address.
    #pragma unroll
    for (int pass = 0; pass < 2; ++pass) {
        const uint32_t byteoff = (uint32_t)(pass * NTHREADS + tid) * 16u;
        const uint32_t lds_u   = lds_base + byteoff;
        const uint32_t lds_v   = lds_base + (uint32_t)(ROW_N * 4) + byteoff;
        asm volatile("global_load_async_to_lds_b128 %0, %1, %2 offset:0"
                     :: "v"(lds_u), "v"(byteoff), "s"(urow) : "memory");
        asm volatile("global_load_async_to_lds_b128 %0, %1, %2 offset:0"
                     :: "v"(lds_v), "v"(byteoff), "s"(vrow) : "memory");
    }
    // Wait for this wave's async DMAs to land in LDS, then cross-wave barrier.
    asm volatile("s_wait_asynccnt 0" ::: "memory");
    __syncthreads();

    // Bitonic sort of both 8192-element arrays concurrently.
    // Per (k,j) stage: N/2 = 4096 pairs per array, 8192 pairs total across both
    // arrays -> 8 compare-exchanges per thread per stage; 91 stages.
    for (int k = 2; k <= ROW_N; k <<= 1) {
        for (int j = k >> 1; j > 0; j >>= 1) {
            #pragma unroll
            for (int p = tid; p < ROW_N; p += NTHREADS) {
                const int arrsel = p >> 12;               // 0 -> u half, 1 -> v half
                const int q      = p & (ROW_N / 2 - 1);   // pair index within array
                // insert a 0 bit at position log2(j): enumerates exactly N/2 pairs
                const int i       = ((q & ~(j - 1)) << 1) | (q & (j - 1));
                const int partner = i | j;
                float* arr = smem + arrsel * ROW_N;
                const bool up = ((i & k) == 0);
                const float x = arr[i];
                const float y = arr[partner];
                if ((x > y) == up) { arr[i] = y; arr[partner] = x; }
            }
            __syncthreads();
        }
    }

    // Per-thread partial sum of |u_sorted - v_sorted|.
    float s = 0.0f;
    #pragma unroll
    for (int i = tid; i < ROW_N; i += NTHREADS)
        s += fabsf(smem[i] - smem[i + ROW_N]);

    // wave32 reduction
    #pragma unroll
    for (int o = 16; o > 0; o >>= 1)
        s += __shfl_down(s, o, 32);

    // All reads of smem are done only after every thread finished its partial
    // sum; barrier, then reuse smem[0..31] for the 32 wave partials.
    __syncthreads();
    if ((tid & 31) == 0) smem[tid >> 5] = s;
    __syncthreads();

    if (tid < 32) {
        float t = smem[tid];
        #pragma unroll
        for (int o = 16; o > 0; o >>= 1)
            t += __shfl_down(t, o, 32);
        if (tid == 0)
            out[row] = t * (1.0f / (float)ROW_N);
    }
}

extern "C" void kernel_launch(void* const* d_in, const int* in_sizes, int n_in,
                              void* d_out, int out_size, void* d_ws, size_t ws_size,
                              hipStream_t stream)
{
    (void)n_in; (void)d_ws; (void)ws_size;
    const float* u = (const float*)d_in[0];
    const float* v = (const float*)d_in[1];
    float* out = (float*)d_out;

    const int B = in_sizes[0] / ROW_N;   // 1024 rows
    (void)out_size;

    WassersteinMetric_14998025798407_kernel<<<dim3(B), dim3(NTHREADS), 0, stream>>>(u, v, out);
}